// SignedGraphConvolutionalNetwork_91190745628896
// MI455X (gfx1250) — compile-verified
//
#include <hip/hip_runtime.h>
#include <math.h>

#define NNODES 8000
#define DIN    2048
#define NEDGE  100000
#define HDIM   64

typedef __attribute__((ext_vector_type(2))) float v2f;
typedef __attribute__((ext_vector_type(8))) float v8f;

__device__ __forceinline__ v8f wmma4(v2f a, v2f b, v8f c) {
  // V_WMMA_F32_16X16X4_F32: D = A(16x4) x B(4x16) + C, full fp32
  return __builtin_amdgcn_wmma_f32_16x16x4_f32(false, a, false, b, (short)0, c, false, false);
}

__global__ void k_zero(float* p, int n) {
  int i = blockIdx.x * blockDim.x + threadIdx.x;
  int stride = gridDim.x * blockDim.x;
  for (; i < n; i += stride) p[i] = 0.f;
}

__global__ void k_count(const int* __restrict__ edges, int e, float* cnt) {
  int i = blockIdx.x * blockDim.x + threadIdx.x;
  if (i < e) atomicAdd(&cnt[edges[i]], 1.0f);
}

// P1 = X @ Wp[0:2048], P2 = X @ Wp[2048:], Q1 = X @ Wn[0:2048], Q2 = X @ Wn[2048:]
// One pass over X produces all 4 projections (both signs, agg-path and self-path).
__global__ void k_gemm_base(const float* __restrict__ X,
                            const float* __restrict__ Wp,
                            const float* __restrict__ Wn,
                            float* __restrict__ P1, float* __restrict__ P2,
                            float* __restrict__ Q1, float* __restrict__ Q2) {
  int lane = threadIdx.x & 31;
  int wave = threadIdx.x >> 5;
  int l15  = lane & 15;
  int hi   = lane >> 4;      // 0 or 1
  int kh   = hi * 2;
  int m0 = blockIdx.x * 16;
  int n0 = wave * 16;
  int row = m0 + l15;
  int col = n0 + l15;
  const float* Ar = X + (size_t)row * DIN + kh;
  v8f p1 = {}, p2 = {}, q1 = {}, q2 = {};
  for (int k = 0; k < DIN; k += 4) {
    v2f a = *reinterpret_cast<const v2f*>(Ar + k);
    int kk = k + kh;
    v2f bp1 = { Wp[(size_t)kk * HDIM + col],           Wp[(size_t)(kk + 1) * HDIM + col] };
    v2f bp2 = { Wp[(size_t)(kk + DIN) * HDIM + col],   Wp[(size_t)(kk + DIN + 1) * HDIM + col] };
    v2f bq1 = { Wn[(size_t)kk * HDIM + col],           Wn[(size_t)(kk + 1) * HDIM + col] };
    v2f bq2 = { Wn[(size_t)(kk + DIN) * HDIM + col],   Wn[(size_t)(kk + DIN + 1) * HDIM + col] };
    p1 = wmma4(a, bp1, p1);
    p2 = wmma4(a, bp2, p2);
    q1 = wmma4(a, bq1, q1);
    q2 = wmma4(a, bq2, q2);
  }
  for (int r = 0; r < 8; ++r) {
    int orow = m0 + r + hi * 8;
    int idx = orow * HDIM + col;
    P1[idx] = p1[r]; P2[idx] = p2[r];
    Q1[idx] = q1[r]; Q2[idx] = q2[r];
  }
}

// dst[row] += src[col] over all edges, 64-dim rows
__global__ void k_scatter64(const int* __restrict__ edges, int e,
                            const float* __restrict__ src, float* __restrict__ dst) {
  int i = blockIdx.x * blockDim.x + threadIdx.x;
  int total = e * HDIM;
  int stride = gridDim.x * blockDim.x;
  for (; i < total; i += stride) {
    int ed = i >> 6, c = i & 63;
    int r = edges[ed];
    int s = edges[e + ed];
    atomicAdd(&dst[r * HDIM + c], src[s * HDIM + c]);
  }
}

// out = tanh(l2norm(scatter_sum/max(cnt,1) + self_proj + bias))
__global__ void k_base_norm(const float* __restrict__ t, const float* __restrict__ cnt,
                            const float* __restrict__ self, const float* __restrict__ bias,
                            float* __restrict__ out) {
  __shared__ float red[HDIM];
  int n = blockIdx.x, c = threadIdx.x;
  float v = t[n * HDIM + c] / fmaxf(cnt[n], 1.0f) + self[n * HDIM + c] + bias[c];
  red[c] = v * v;
  __syncthreads();
  for (int s = 32; s > 0; s >>= 1) {
    if (c < s) red[c] += red[c + s];
    __syncthreads();
  }
  float nrm = fmaxf(sqrtf(red[0]), 1e-12f);
  out[n * HDIM + c] = tanhf(v / nrm);
}

// a = (a + self) / (cnt + 1)   (self-loop mean for the deep layer)
__global__ void k_deep_mean(float* a, const float* __restrict__ self,
                            const float* __restrict__ cnt, int total) {
  int i = blockIdx.x * blockDim.x + threadIdx.x;
  if (i < total) a[i] = (a[i] + self[i]) / (cnt[i >> 6] + 1.0f);
}

// out = tanh(l2norm([A0|A1|A2] @ W(192x64) + bias)), fused via LDS
__global__ void k_gemm_deep(const float* __restrict__ A0, const float* __restrict__ A1,
                            const float* __restrict__ A2, const float* __restrict__ W,
                            const float* __restrict__ bias, float* __restrict__ out) {
  __shared__ float tile[16][HDIM];
  __shared__ float nrm[16];
  int lane = threadIdx.x & 31;
  int wave = threadIdx.x >> 5;
  int l15  = lane & 15;
  int hi   = lane >> 4;
  int kh   = hi * 2;
  int m0 = blockIdx.x * 16;
  int n0 = wave * 16;
  int row = m0 + l15;
  int col = n0 + l15;
  v8f acc = {};
  const float* As[3] = { A0, A1, A2 };
  for (int s = 0; s < 3; ++s) {
    const float* Ar = As[s] + row * HDIM + kh;
    const float* Wc = W + (size_t)(s * HDIM) * HDIM + col;
    for (int k = 0; k < HDIM; k += 4) {
      v2f a = *reinterpret_cast<const v2f*>(Ar + k);
      int kk = k + kh;
      v2f b = { Wc[kk * HDIM], Wc[(kk + 1) * HDIM] };
      acc = wmma4(a, b, acc);
    }
  }
  float bcol = bias[col];
  for (int r = 0; r < 8; ++r)
    tile[r + hi * 8][col] = acc[r] + bcol;
  __syncthreads();
  if (threadIdx.x < 16) {
    float s = 0.f;
    for (int c = 0; c < HDIM; ++c) { float v = tile[threadIdx.x][c]; s += v * v; }
    nrm[threadIdx.x] = fmaxf(sqrtf(s), 1e-12f);
  }
  __syncthreads();
  for (int j = 0; j < 8; ++j) {
    int idx = threadIdx.x * 8 + j;
    int r = idx >> 6, c = idx & 63;
    out[(m0 + r) * HDIM + c] = tanhf(tile[r][c] / nrm[r]);
  }
}

// X_mol = l2norm([h_pos2 | h_neg2]); write both aligned ws copy and d_out slot
__global__ void k_xmol(const float* __restrict__ hp, const float* __restrict__ hn,
                       float* __restrict__ xmol, float* __restrict__ out_xmol) {
  __shared__ float red[128];
  int n = blockIdx.x, t = threadIdx.x;
  float v = (t < HDIM) ? hp[n * HDIM + t] : hn[n * HDIM + (t - HDIM)];
  red[t] = v * v;
  __syncthreads();
  for (int s = 64; s > 0; s >>= 1) {
    if (t < s) red[t] += red[t + s];
    __syncthreads();
  }
  float nrm = fmaxf(sqrtf(red[0]), 1e-12f);
  float o = v / nrm;
  xmol[n * 128 + t] = o;
  out_xmol[n * 128 + t] = o;
}

// pred = (Xm @ Xm^T) * mask, fused MSE partial reduction. Block = 8 waves = 32x64 tile.
__global__ void k_pred(const float* __restrict__ Xm, const float* __restrict__ mask,
                       const float* __restrict__ labels, float* __restrict__ pred,
                       float* __restrict__ lossAcc) {
  __shared__ float red[256];
  int lane = threadIdx.x & 31;
  int wave = threadIdx.x >> 5;
  int l15  = lane & 15;
  int hi   = lane >> 4;
  int kh   = hi * 2;
  int m0 = blockIdx.y * 32 + (wave >> 2) * 16;
  int n0 = blockIdx.x * 64 + (wave & 3) * 16;
  int row = m0 + l15;
  int col = n0 + l15;
  const float* Ar = Xm + row * 128 + kh;
  const float* Br = Xm + col * 128 + kh;   // B[k,n] = Xm[n,k] -> contiguous in k
  v8f acc = {};
  for (int k = 0; k < 128; k += 4) {
    v2f a = *reinterpret_cast<const v2f*>(Ar + k);
    v2f b = *reinterpret_cast<const v2f*>(Br + k);
    acc = wmma4(a, b, acc);
  }
  float ls = 0.f;
  for (int r = 0; r < 8; ++r) {
    int orow = m0 + r + hi * 8;
    int idx = orow * NNODES + col;
    float p = acc[r] * mask[idx];
    pred[idx] = p;
    float d = p - labels[idx];
    ls += d * d;
  }
  red[threadIdx.x] = ls;
  __syncthreads();
  for (int s = 128; s > 0; s >>= 1) {
    if (threadIdx.x < s) red[threadIdx.x] += red[threadIdx.x + s];
    __syncthreads();
  }
  if (threadIdx.x == 0) atomicAdd(lossAcc, red[0]);
}

__global__ void k_loss(const float* __restrict__ acc, float* __restrict__ out) {
  out[0] = acc[0] * (1.0f / ((float)NNODES * (float)NNODES));
}

extern "C" void kernel_launch(void* const* d_in, const int* in_sizes, int n_in,
                              void* d_out, int out_size, void* d_ws, size_t ws_size,
                              hipStream_t stream) {
  const float* X     = (const float*)d_in[0];
  const int*   e_pos = (const int*)  d_in[1];
  const int*   e_neg = (const int*)  d_in[2];
  const float* labels= (const float*)d_in[3];
  const float* mask  = (const float*)d_in[4];
  const float* W_pb  = (const float*)d_in[5];
  const float* b_pb  = (const float*)d_in[6];
  const float* W_nb  = (const float*)d_in[7];
  const float* b_nb  = (const float*)d_in[8];
  const float* W_pd  = (const float*)d_in[9];
  const float* b_pd  = (const float*)d_in[10];
  const float* W_nd  = (const float*)d_in[11];
  const float* b_nd  = (const float*)d_in[12];

  float* out = (float*)d_out;
  float* out_xmol = out + 1;
  float* out_pred = out + 1 + NNODES * 128;

  const int NH = NNODES * HDIM;
  float* ws = (float*)d_ws;
  float* cnt_pos = ws;                 // N
  float* cnt_neg = cnt_pos + NNODES;   // N
  float* t_pos   = cnt_neg + NNODES;   // NH
  float* t_neg   = t_pos + NH;
  float* a_pp    = t_neg + NH;         // deep scatter sums (4x NH)
  float* a_nn    = a_pp + NH;
  float* a_pn    = a_nn + NH;
  float* a_np    = a_pn + NH;
  float* P1      = a_np + NH;          // base projections (4x NH)
  float* P2      = P1 + NH;
  float* Q1      = P2 + NH;
  float* Q2      = Q1 + NH;
  float* h_pos   = Q2 + NH;
  float* h_neg   = h_pos + NH;
  float* h_pos2  = h_neg + NH;
  float* h_neg2  = h_pos2 + NH;
  float* xmol    = h_neg2 + NH;        // N*128 (8B-aligned copy for GEMM)
  float* lossAcc = xmol + NNODES * 128;

  // zero: counts + scatter accumulators + loss scalar
  int zn = 2 * NNODES + 6 * NH;
  k_zero<<<2048, 256, 0, stream>>>(cnt_pos, zn);
  k_zero<<<1, 32, 0, stream>>>(lossAcc, 1);

  // degree counts
  k_count<<<(NEDGE + 255) / 256, 256, 0, stream>>>(e_pos, NEDGE, cnt_pos);
  k_count<<<(NEDGE + 255) / 256, 256, 0, stream>>>(e_neg, NEDGE, cnt_neg);

  // base projections: one pass over X, 4 outputs (WMMA f32 16x16x4, K=2048)
  k_gemm_base<<<NNODES / 16, 128, 0, stream>>>(X, W_pb, W_nb, P1, P2, Q1, Q2);

  // scatter-mean in projected (64-dim) space
  int sgrid = (NEDGE * HDIM + 255) / 256;
  k_scatter64<<<sgrid, 256, 0, stream>>>(e_pos, NEDGE, P1, t_pos);
  k_scatter64<<<sgrid, 256, 0, stream>>>(e_neg, NEDGE, Q1, t_neg);

  // base layer output: mean + self + bias, l2norm, tanh
  k_base_norm<<<NNODES, HDIM, 0, stream>>>(t_pos, cnt_pos, P2, b_pb, h_pos);
  k_base_norm<<<NNODES, HDIM, 0, stream>>>(t_neg, cnt_neg, Q2, b_nb, h_neg);

  // deep scatter sums (both signs of edges x both hidden states)
  k_scatter64<<<sgrid, 256, 0, stream>>>(e_pos, NEDGE, h_pos, a_pp);
  k_scatter64<<<sgrid, 256, 0, stream>>>(e_pos, NEDGE, h_neg, a_pn);
  k_scatter64<<<sgrid, 256, 0, stream>>>(e_neg, NEDGE, h_neg, a_nn);
  k_scatter64<<<sgrid, 256, 0, stream>>>(e_neg, NEDGE, h_pos, a_np);

  // self-loop means
  int dgrid = (NH + 255) / 256;
  k_deep_mean<<<dgrid, 256, 0, stream>>>(a_pp, h_pos, cnt_pos, NH);
  k_deep_mean<<<dgrid, 256, 0, stream>>>(a_nn, h_neg, cnt_neg, NH);
  k_deep_mean<<<dgrid, 256, 0, stream>>>(a_pn, h_neg, cnt_pos, NH);
  k_deep_mean<<<dgrid, 256, 0, stream>>>(a_np, h_pos, cnt_neg, NH);

  // deep convs (K=192 WMMA GEMM + fused bias/l2norm/tanh)
  k_gemm_deep<<<NNODES / 16, 128, 0, stream>>>(a_pp, a_nn, h_pos, W_pd, b_pd, h_pos2);
  k_gemm_deep<<<NNODES / 16, 128, 0, stream>>>(a_pn, a_np, h_neg, W_nd, b_nd, h_neg2);

  // X_mol (writes aligned ws copy + d_out region)
  k_xmol<<<NNODES, 128, 0, stream>>>(h_pos2, h_neg2, xmol, out_xmol);

  // pairwise GEMM + mask + MSE
  dim3 pgrid(NNODES / 64, NNODES / 32);
  k_pred<<<pgrid, 256, 0, stream>>>(xmol, mask, labels, out_pred, lossAcc);

  // loss
  k_loss<<<1, 1, 0, stream>>>(lossAcc, out);
}